// BertModelWrapper_noise_44418551775571
// MI455X (gfx1250) — compile-verified
//
#include <hip/hip_runtime.h>

#define H      768
#define SLEN   512
#define BATCH  32
#define NTOK   (BATCH * SLEN)
#define HALFN  6291456u          /* (32*512*768)/2 */
#define NOISE_STD 0.5f
#define LN_EPS 1e-12f

typedef __attribute__((ext_vector_type(2))) float v2f;
typedef __attribute__((ext_vector_type(8))) float v8f;

__device__ __forceinline__ unsigned rotl32(unsigned x, unsigned r) {
  return (x << r) | (x >> (32u - r));
}

// Broadcast lane 0's value to all lanes via SALU (EXEC all-ones here),
// avoiding the ds_bpermute round trip that __shfl(x,0,32) emits.
__device__ __forceinline__ float bcast0_f32(float x) {
  return __int_as_float(__builtin_amdgcn_readfirstlane(__float_as_int(x)));
}

// JAX threefry2x32, key = jax.random.key(42) -> (k0,k1) = (0,42).
// Counter pair (p, p+HALFN); sel=0 returns x0 (first half of flat array),
// sel=1 returns x1 (second half). Matches jax._src.prng.threefry2x32.
__device__ __forceinline__ unsigned threefry_bits(unsigned p, int sel) {
  const unsigned ks0 = 0u;
  const unsigned ks1 = 42u;
  const unsigned ks2 = 0u ^ 42u ^ 0x1BD11BDAu;
  unsigned x0 = p + ks0;
  unsigned x1 = (p + HALFN) + ks1;
#define TF_ROUND(r) { x0 += x1; x1 = rotl32(x1, r); x1 ^= x0; }
  TF_ROUND(13) TF_ROUND(15) TF_ROUND(26) TF_ROUND(6)   // i=0
  x0 += ks1; x1 += ks2 + 1u;
  TF_ROUND(17) TF_ROUND(29) TF_ROUND(16) TF_ROUND(24)  // i=1
  x0 += ks2; x1 += ks0 + 2u;
  TF_ROUND(13) TF_ROUND(15) TF_ROUND(26) TF_ROUND(6)   // i=2
  x0 += ks0; x1 += ks1 + 3u;
  TF_ROUND(17) TF_ROUND(29) TF_ROUND(16) TF_ROUND(24)  // i=3
  x0 += ks1; x1 += ks2 + 4u;
  TF_ROUND(13) TF_ROUND(15) TF_ROUND(26) TF_ROUND(6)   // i=4
  x0 += ks2; x1 += ks0 + 5u;
#undef TF_ROUND
  return sel ? x1 : x0;
}

// Single-precision erfinv (M. Giles, 2012) — matches XLA's erf_inv closely.
__device__ __forceinline__ float erfinv_f(float x) {
  float w = -__logf(fmaf(-x, x, 1.0f));   // -log((1-x)(1+x))
  float p;
  if (w < 5.0f) {
    w -= 2.5f;
    p = 2.81022636e-08f;
    p = fmaf(p, w, 3.43273939e-07f);
    p = fmaf(p, w, -3.5233877e-06f);
    p = fmaf(p, w, -4.39150654e-06f);
    p = fmaf(p, w, 0.00021858087f);
    p = fmaf(p, w, -0.00125372503f);
    p = fmaf(p, w, -0.00417768164f);
    p = fmaf(p, w, 0.246640727f);
    p = fmaf(p, w, 1.50140941f);
  } else {
    w = __fsqrt_rn(w) - 3.0f;
    p = -0.000200214257f;
    p = fmaf(p, w, 0.000100950558f);
    p = fmaf(p, w, 0.00134934322f);
    p = fmaf(p, w, -0.00367342844f);
    p = fmaf(p, w, 0.00573950773f);
    p = fmaf(p, w, -0.0076224613f);
    p = fmaf(p, w, 0.00943887047f);
    p = fmaf(p, w, 1.00167406f);
    p = fmaf(p, w, 2.83297682f);
  }
  return p * x;
}

// noise[f] = jax.random.normal(key(42), [B,S,H])[f] * STD
__device__ __forceinline__ float jax_noise(unsigned f) {
  const int sel = f >= HALFN;
  const unsigned p = sel ? f - HALFN : f;
  const unsigned bits = threefry_bits(p, sel);
  const float u01 = __uint_as_float((bits >> 9) | 0x3f800000u) - 1.0f; // [0,1)
  const float lo = -0.99999994f;            // nextafter(-1,0)
  float u = fmaxf(lo, fmaf(u01, 2.0f, lo)); // (maxval-minval) rounds to 2.0f
  return 1.41421356237f * erfinv_f(u) * NOISE_STD;
}

// --- kernel 1: per-sequence importance stats: max, min(inf->99), sum -------
__global__ __launch_bounds__(256) void row_stats_kernel(
    const float* __restrict__ imp, float* __restrict__ stats) {
  const int b = blockIdx.x;
  const int tid = threadIdx.x;
  const int lane = tid & 31, wave = tid >> 5;
  const float v0 = imp[b * SLEN + tid];
  const float v1 = imp[b * SLEN + 256 + tid];
  float mx = fmaxf(v0, v1);
  const float t0 = __builtin_isinf(v0) ? 99.0f : v0;
  const float t1 = __builtin_isinf(v1) ? 99.0f : v1;
  float mn = fminf(t0, t1);
  float sm = v0 + v1;
#pragma unroll
  for (int off = 16; off > 0; off >>= 1) {
    mx = fmaxf(mx, __shfl_xor(mx, off, 32));
    mn = fminf(mn, __shfl_xor(mn, off, 32));
    sm += __shfl_xor(sm, off, 32);
  }
  __shared__ float smx[8], smn[8], ssm[8];
  if (lane == 0) { smx[wave] = mx; smn[wave] = mn; ssm[wave] = sm; }
  __syncthreads();
  if (tid == 0) {
    float rmx = smx[0], rmn = smn[0], rsm = ssm[0];
#pragma unroll
    for (int w = 1; w < 8; ++w) {
      rmx = fmaxf(rmx, smx[w]); rmn = fminf(rmn, smn[w]); rsm += ssm[w];
    }
    stats[b] = rmx; stats[32 + b] = rmn; stats[64 + b] = rsm;
  }
}

// --- kernel 2: global score sum (for the sum()==0 branch) ------------------
__global__ void total_sum_kernel(float* __restrict__ stats) {
  float v = stats[64 + threadIdx.x];
#pragma unroll
  for (int off = 16; off > 0; off >>= 1) v += __shfl_xor(v, off, 32);
  if (threadIdx.x == 0) stats[96] = v;
}

// --- kernel 3: fused embed + LayerNorm + threefry noise --------------------
// One block (8 wave32) per token; cross-wave LN reduction via f32 WMMA.
__global__ __launch_bounds__(256) void bert_noise_main(
    const int* __restrict__ input_ids,
    const int* __restrict__ token_type_ids,
    const float* __restrict__ importance,
    const float* __restrict__ word_emb,
    const float* __restrict__ pos_emb,
    const float* __restrict__ type_emb,
    const float* __restrict__ gamma,
    const float* __restrict__ beta,
    const float* __restrict__ stats,
    float* __restrict__ out) {
  const int token = blockIdx.x;           // b*S + s
  const int b = token >> 9;
  const int s = token & (SLEN - 1);
  const int tid = threadIdx.x;
  const int lane = tid & 31;
  const int wave = tid >> 5;

  const int wid = input_ids[token];
  const int tt  = token_type_ids[token];
  const float* __restrict__ wrow = word_emb + (size_t)wid * H;
  const float* __restrict__ prow = pos_emb + (size_t)s * H;
  const float* __restrict__ trow = type_emb + (size_t)tt * H;

  float e[3];
  float sum = 0.0f, sq = 0.0f;
#pragma unroll
  for (int i = 0; i < 3; ++i) {           // coalesced 128B/wave per table
    const int h = tid + i * 256;
    const float v = wrow[h] + prow[h] + trow[h];
    e[i] = v;
    sum += v;
    sq = fmaf(v, v, sq);
  }

  // intra-wave tree reduction (wave32)
#pragma unroll
  for (int off = 16; off > 0; off >>= 1) {
    sum += __shfl_xor(sum, off, 32);
    sq  += __shfl_xor(sq,  off, 32);
  }
  __shared__ float part[16];              // [0..7]=wave sums, [8..15]=wave sq
  if (lane == 0) { part[wave] = sum; part[8 + wave] = sq; }
  __syncthreads();

  // Cross-wave combine with one V_WMMA_F32_16X16X4_F32 per wave (full EXEC):
  // A(16x4) rows 0-1 = 8 wave sums, rows 2-3 = 8 wave sum-sqs; B = ones(4x16).
  // ISA A layout: VGPR0 holds K=0 (lanes 0-15) / K=2 (lanes 16-31), VGPR1 K=1/K=3.
  const int m  = lane & 15;
  const int hi = lane >> 4;
  float a0 = 0.0f, a1 = 0.0f;
  if (m < 4) {
    a0 = part[4 * m + 2 * hi];
    a1 = part[4 * m + 2 * hi + 1];
  }
  v2f A;  A[0] = a0;   A[1] = a1;
  v2f Bv; Bv[0] = 1.0f; Bv[1] = 1.0f;     // all-ones B: layout-invariant
  v8f C = {};
  v8f D = __builtin_amdgcn_wmma_f32_16x16x4_f32(
      /*neg_a=*/false, A, /*neg_b=*/false, Bv,
      /*c_mod=*/(short)0, C, /*reuse_a=*/false, /*reuse_b=*/false);
  // Lane 0: D[0]=sum(waves0-3), D[1]=sum(waves4-7), D[2]/D[3]=same for x^2.
  const float totSum = bcast0_f32(D[0] + D[1]);
  const float totSq  = bcast0_f32(D[2] + D[3]);

  const float mean = totSum * (1.0f / H);
  float var = fmaf(totSq, 1.0f / H, -mean * mean);
  var = fmaxf(var, 0.0f);
  const float rstd = rsqrtf(var + LN_EPS);

  // sigma (uniform per token)
  const float sc   = importance[token];
  const float smax = stats[b];
  const float smin = stats[32 + b];
  const float tot  = stats[96];
  float norm = (sc - smin) / (smax - smin);
  if (tot == 0.0f) norm = sc;
  const bool fin = __builtin_isfinite(sc);
  float score = fin ? norm : 0.0f;
  bool mask = fin;
  if (s == 0) { score = 1.0f; mask = true; }
  const float sigma = mask ? (1.0f - score) : 0.0f;

  const unsigned base = (unsigned)token * H;
#pragma unroll
  for (int i = 0; i < 3; ++i) {
    const int h = tid + i * 256;
    const float y = fmaf((e[i] - mean) * rstd, gamma[h], beta[h]);
    const float nz = jax_noise(base + (unsigned)h);
    out[base + h] = fmaf(nz * sigma, y, y);   // y * (1 + noise*sigma)
  }
}

extern "C" void kernel_launch(void* const* d_in, const int* in_sizes, int n_in,
                              void* d_out, int out_size, void* d_ws, size_t ws_size,
                              hipStream_t stream) {
  const int*   input_ids      = (const int*)d_in[0];
  const int*   token_type_ids = (const int*)d_in[1];
  const float* importance     = (const float*)d_in[2];
  const float* word_emb       = (const float*)d_in[3];
  const float* pos_emb        = (const float*)d_in[4];
  const float* type_emb       = (const float*)d_in[5];
  const float* ln_gamma       = (const float*)d_in[6];
  const float* ln_beta        = (const float*)d_in[7];
  float* out   = (float*)d_out;
  float* stats = (float*)d_ws;   // [0..31] max, [32..63] min, [64..95] rowsum, [96] total

  hipLaunchKernelGGL(row_stats_kernel, dim3(BATCH), dim3(256), 0, stream,
                     importance, stats);
  hipLaunchKernelGGL(total_sum_kernel, dim3(1), dim3(32), 0, stream, stats);
  hipLaunchKernelGGL(bert_noise_main, dim3(NTOK), dim3(256), 0, stream,
                     input_ids, token_type_ids, importance, word_emb, pos_emb,
                     type_emb, ln_gamma, ln_beta, stats, out);
}